// AttentionMechanism_266287972502
// MI455X (gfx1250) — compile-verified
//
#include <hip/hip_runtime.h>
#include <hip/hip_bf16.h>

typedef float v2f __attribute__((ext_vector_type(2)));
typedef float v4f __attribute__((ext_vector_type(4)));
typedef float v8f __attribute__((ext_vector_type(8)));

#define LRELU_ALPHA 0.2f

// ---------------------------------------------------------------------------
// Phase 1: dual mat-vec  s_src = X @ a_src, s_dst = X @ a_dst
// via V_WMMA_F32_16X16X4_F32. One wave handles 16 rows of X; B-matrix column 0
// carries a_src, column 1 carries a_dst (other 14 columns are zero).
//
// f32 A-operand layout (16x4, 2 VGPRs/lane):
//   lanes 0-15 : M = lane,    VGPR0 = K=k0+0, VGPR1 = K=k0+1
//   lanes 16-31: M = lane-16, VGPR0 = K=k0+2, VGPR1 = K=k0+3
// B-operand (4x16) mirrors this with N = lane&15 striped across lanes.
// D/C f32 16x16: lanes 0-15 hold N=lane, M = vgpr(+0); lanes 16-31 M = vgpr+8.
// ---------------------------------------------------------------------------
__global__ void __launch_bounds__(256)
gat_dual_matvec_wmma(const float* __restrict__ X,
                     const float* __restrict__ att,
                     float* __restrict__ s_src,
                     float* __restrict__ s_dst,
                     int N, int F)
{
    extern __shared__ float sa[];          // 2*F floats: [a_src | a_dst]
    for (int t = threadIdx.x; t < 2 * F; t += blockDim.x) sa[t] = att[t];
    __syncthreads();

    const int wave     = (int)((blockIdx.x * blockDim.x + threadIdx.x) >> 5);
    const int lane     = (int)(threadIdx.x & 31);
    const int row_base = wave * 16;
    if (row_base >= N) return;             // uniform per wave: EXEC stays all-1s

    const int n     = lane & 15;           // A: row within tile / B: column
    const int khalf = (lane >> 4) << 1;    // 0 for lanes 0-15, 2 for lanes 16-31
    const float* __restrict__ Xrow = X + (size_t)(row_base + n) * (size_t)F;

    v8f c = {};                            // 16x16 f32 accumulator (8 VGPRs)
    for (int k0 = 0; k0 < F; k0 += 4) {
        v2f a;
        a.x = Xrow[k0 + khalf];
        a.y = Xrow[k0 + khalf + 1];

        v2f b = { 0.0f, 0.0f };
        if (n < 2) {                       // col 0 = a_src, col 1 = a_dst
            const float* av = sa + n * F;
            b.x = av[k0 + khalf];
            b.y = av[k0 + khalf + 1];
        }
        // (neg_a, A, neg_b, B, c_mod, C, reuse_a, reuse_b)
        c = __builtin_amdgcn_wmma_f32_16x16x4_f32(
                false, a, false, b, (short)0, c, false, false);
    }

    // Column 0 of D = s_src for rows row_base..row_base+15, column 1 = s_dst.
    if (n < 2) {
        float* outp = (n == 0) ? s_src : s_dst;
        const int r0 = row_base + ((lane >> 4) << 3);   // +0 or +8
        #pragma unroll
        for (int i = 0; i < 8; ++i) outp[r0 + i] = c[i];
    }
}

// ---------------------------------------------------------------------------
// Phase 2: out[i][j] = leaky_relu(s_src[i] + s_dst[j]).
// Store-bandwidth bound (256 MB > 192 MB L2): non-temporal B128 stores.
// Thread tile = 4 rows x 4 cols (64 B of stores). s_src[i] is wave-uniform
// (blockIdx.y) -> scalar loads; s_dst float4 is reused across the 4 rows.
// ---------------------------------------------------------------------------
__global__ void __launch_bounds__(256)
gat_outer_lrelu(const float* __restrict__ s_src,
                const float* __restrict__ s_dst,
                float* __restrict__ out,
                int N)
{
    const int j = (int)(blockIdx.x * blockDim.x + threadIdx.x) * 4;
    if (j >= N) return;
    const int i0 = (int)blockIdx.y * 4;

    const v4f sd = *(const v4f*)(s_dst + j);

    #pragma unroll
    for (int r = 0; r < 4; ++r) {
        const int i = i0 + r;
        if (i >= N) break;
        const float si = s_src[i];         // uniform -> s_load
        v4f v;
        v.x = si + sd.x;
        v.y = si + sd.y;
        v.z = si + sd.z;
        v.w = si + sd.w;
        v4f o;
        o.x = (v.x > 0.0f) ? v.x : v.x * LRELU_ALPHA;
        o.y = (v.y > 0.0f) ? v.y : v.y * LRELU_ALPHA;
        o.z = (v.z > 0.0f) ? v.z : v.z * LRELU_ALPHA;
        o.w = (v.w > 0.0f) ? v.w : v.w * LRELU_ALPHA;
        __builtin_nontemporal_store(o, (v4f*)(out + (size_t)i * (size_t)N + j));
    }
}

// ---------------------------------------------------------------------------
extern "C" void kernel_launch(void* const* d_in, const int* in_sizes, int n_in,
                              void* d_out, int out_size, void* d_ws, size_t ws_size,
                              hipStream_t stream)
{
    const float* X   = (const float*)d_in[0];   // (N, F) f32
    const float* att = (const float*)d_in[1];   // (2F, 1) f32

    const int F = in_sizes[1] / 2;              // 256
    const int N = in_sizes[0] / F;              // 8192

    float* s_src = (float*)d_ws;                // N floats
    float* s_dst = s_src + N;                   // N floats (64 KB total)

    // Phase 1: 1 wave per 16 rows, 8 waves per 256-thread block.
    const int waves   = (N + 15) / 16;
    const int blocks1 = (waves + 7) / 8;
    gat_dual_matvec_wmma<<<blocks1, 256, 2 * F * (int)sizeof(float), stream>>>(
        X, att, s_src, s_dst, N, F);

    // Phase 2: threads cover N/4 column-quads, blockIdx.y covers N/4 row-quads.
    dim3 grid2((unsigned)((N / 4 + 255) / 256), (unsigned)((N + 3) / 4));
    gat_outer_lrelu<<<grid2, 256, 0, stream>>>(s_src, s_dst, (float*)d_out, N);
}